// ContourletCNN_77824807404019
// MI455X (gfx1250) — compile-verified
//
#include <hip/hip_runtime.h>

// ---------------------------------------------------------------------------
// Contourlet forward (3-level Laplacian pyramid + 8-band DFB/level + 2D DWT)
//   in : x (8,16,512,512) fp32
//   out: [coarse(8,64,32,32) | fine2(8,128,64,64) | fine1(8,128,128,128)
//         | fine0(8,128,256,256)]  fp32 flat
// Bandwidth-bound (~1.4 GB traffic, ~2 GFLOP): fp32 throughout, fused
// shear+butterfly DFB stages, fused LP high-pass, and the LP low-pass done
// on the matrix pipe: xlo 16x16 tile = banded M(16x35) x T(35x16) as nine
// chained V_WMMA_F32_16X16X4_F32 per wave (T staged in LDS, tile staged via
// GLOBAL_LOAD_ASYNC_TO_LDS_B32 when the toolchain exposes it).
// Workspace requirement: 2*N0 + N0/2 floats = 335,544,320 bytes in d_ws.
// ---------------------------------------------------------------------------

typedef __attribute__((ext_vector_type(2))) float v2f;
typedef __attribute__((ext_vector_type(8))) float v8f;

#define AS1 __attribute__((address_space(1)))
#define AS3 __attribute__((address_space(3)))

#if defined(__has_builtin)
#if __has_builtin(__builtin_amdgcn_global_load_async_to_lds_b32)
#define HAVE_ASYNC_LDS 1
#endif
#endif
#ifndef HAVE_ASYNC_LDS
#define HAVE_ASYNC_LDS 0
#endif

// maxflat pyramid filters (exact closed forms, see reference)
#define H0 0.125f
#define H1 0.35355339059f
#define H2c 1.25f
#define G0 0.02588834765f
#define G1 -0.07322330470f
#define G2 -0.06878156650f
#define G3 0.85355339059f

__constant__ float c_h[5]  = {H0, H1, H2c, H1, H0};
__constant__ float c_h1[7] = {-G0, G1, -G2, G3, -G2, G1, -G0};

__device__ __forceinline__ int imod(int v, int n) {
  int r = v % n;
  return (r < 0) ? r + n : r;
}

// h[t] for t in [0,4], 0 outside -- branch-free (3 cndmask chains)
__device__ __forceinline__ float hval(int t) {
  float r = (t == 2) ? H2c : 0.f;
  r = (t == 1 || t == 3) ? H1 : r;
  r = (t == 0 || t == 4) ? H0 : r;
  return r;
}

// --------------------------------------------------------------------------
// Laplacian low-pass: xlo(a,b) = sum_{m,n} h[m]h[n] x[(2a-2+m)%H,(2b-2+n)%W]
// One wave per 16x16 xlo tile:
//   1) stage 35x35 wrapped input tile in LDS (async-to-LDS if available)
//   2) column filter -> T[35][16] in LDS
//   3) row filter + decimation = banded 16x35 matrix x T as nine chained
//      V_WMMA_F32_16X16X4_F32; A chunks prebuilt branch-free in registers.
// --------------------------------------------------------------------------
__global__ void k_lplo(const float* __restrict__ x, float* __restrict__ xlo,
                       int H, int W) {
  const int H2 = H >> 1, W2 = W >> 1;
  const int lane = threadIdx.x;                 // 0..31 (wave32)
  const int b0 = blockIdx.x * 16;               // xlo col base
  const int a0 = blockIdx.y * 16;               // xlo row base
  const int img = blockIdx.z;
  const float* xi = x + (size_t)img * H * W;

  __shared__ float S[35][36];                   // raw input tile (padded)
  __shared__ float T[36][16];                   // column-filtered tile

  const int r0 = 2 * a0 - 2, c0 = 2 * b0 - 2;
#if HAVE_ASYNC_LDS
  for (int idx = lane; idx < 35 * 35; idx += 32) {
    int r = idx / 35, cc = idx % 35;
    int gr = r0 + r;  gr += (gr < 0) ? H : 0;  gr -= (gr >= H) ? H : 0;
    int gc = c0 + cc; gc += (gc < 0) ? W : 0;  gc -= (gc >= W) ? W : 0;
    const float* gp = xi + (size_t)gr * W + gc;
    __builtin_amdgcn_global_load_async_to_lds_b32(
        (AS1 int*)(uintptr_t)gp,
        (AS3 int*)(uint32_t)(uintptr_t)&S[r][cc], 0, 0);
  }
#if defined(__has_builtin) && __has_builtin(__builtin_amdgcn_s_wait_asynccnt)
  __builtin_amdgcn_s_wait_asynccnt(0);
#else
  asm volatile("s_wait_asynccnt 0x0" ::: "memory");
#endif
#else
  for (int idx = lane; idx < 35 * 35; idx += 32) {
    int r = idx / 35, cc = idx % 35;
    int gr = r0 + r;  gr += (gr < 0) ? H : 0;  gr -= (gr >= H) ? H : 0;
    int gc = c0 + cc; gc += (gc < 0) ? W : 0;  gc -= (gc >= W) ? W : 0;
    S[r][cc] = xi[(size_t)gr * W + gc];
  }
#endif
  __syncthreads();

  for (int idx = lane; idx < 35 * 16; idx += 32) {
    int r = idx >> 4, bb = idx & 15;
    const float* s = &S[r][2 * bb];
    T[r][bb] = H0 * s[0] + H1 * s[1] + H2c * s[2] + H1 * s[3] + H0 * s[4];
  }
  if (lane < 16) T[35][lane] = 0.f;             // K-padding row
  __syncthreads();

  // Prebuild all nine 16x4 A chunks of band matrix M[a][r] = h[r-2a]
  // (lanes 0-15: K=4c+0/1, row=lane; lanes 16-31: K=4c+2/3, row=lane-16).
  const int row16 = lane & 15;
  const bool hiHalf = lane >= 16;
  float Ax[9], Ay[9];
#pragma unroll
  for (int c4 = 0; c4 < 9; ++c4) {
    const int t0 = 4 * c4 + (hiHalf ? 2 : 0) - 2 * row16;
    Ax[c4] = hval(t0);
    Ay[c4] = hval(t0 + 1);
  }

  v8f acc = {};
#pragma unroll
  for (int c4 = 0; c4 < 9; ++c4) {
    const int k0 = 4 * c4 + (hiHalf ? 2 : 0);
    v2f A, Bv;
    A.x = Ax[c4];
    A.y = Ay[c4];
    Bv.x = T[k0][row16];                        // B 4x16: lane n = column n
    Bv.y = T[k0 + 1][row16];
    acc = __builtin_amdgcn_wmma_f32_16x16x4_f32(
        /*neg_a=*/false, A, /*neg_b=*/false, Bv,
        /*c_mod=*/(short)0, acc, /*reuse_a=*/false, /*reuse_b=*/false);
  }

  // D layout: lanes 0-15 -> rows 0..7 (vgpr v), col=lane; lanes 16-31 ->
  // rows 8..15, col=lane-16.
  float* lo = xlo + (size_t)img * H2 * W2;
  const int rbase = hiHalf ? 8 : 0;
#pragma unroll
  for (int v = 0; v < 8; ++v)
    lo[(size_t)(a0 + rbase + v) * W2 + (b0 + row16)] = acc[v];
}

// --------------------------------------------------------------------------
// Laplacian high-pass, fused:  xhi = x - (g (x) g) * upsample2(xlo)
// Zero-inserted upsample -> only parity-matched taps survive:
//   i odd  -> p in {0,2,4,6} coeffs {G0,G2,G2,G0}
//   i even -> p in {1,3,5}   coeffs {G1,G3,G1} (+ phantom 0-weight tap)
// Fully unrolled 4x4 grid, coefficients via single cndmask selects.
// --------------------------------------------------------------------------
__global__ void k_lphi(const float* __restrict__ x, const float* __restrict__ xlo,
                       float* __restrict__ xhi, int H, int W, int total) {
  int idx = blockIdx.x * blockDim.x + threadIdx.x;
  if (idx >= total) return;
  const int W2 = W >> 1, H2 = H >> 1;
  int j = idx % W;
  int t = idx / W;
  int i = t % H;
  int img = t / H;
  const float* lo = xlo + (size_t)img * H2 * W2;

  if (idx + W < total) __builtin_prefetch(x + idx + W, 0, 1);  // global_prefetch_b8

  const bool iodd = (i & 1) != 0, jodd = (j & 1) != 0;
  const int p0 = iodd ? 0 : 1;
  const int q0 = jodd ? 0 : 1;
  const float cp0 = iodd ? G0 : G1, cp1 = iodd ? G2 : G3;
  const float cp2 = iodd ? G2 : G1, cp3 = iodd ? G0 : 0.f;
  const float cq0 = jodd ? G0 : G1, cq1 = jodd ? G2 : G3;
  const float cq2 = jodd ? G2 : G1, cq3 = jodd ? G0 : 0.f;

  int rr[4], cc[4];
#pragma unroll
  for (int k = 0; k < 4; ++k) {
    int u = i - 3 + p0 + 2 * k; u += (u < 0) ? H : 0; u -= (u >= H) ? H : 0;
    rr[k] = (u >> 1) * W2;
    int v = j - 3 + q0 + 2 * k; v += (v < 0) ? W : 0; v -= (v >= W) ? W : 0;
    cc[k] = v >> 1;
  }
  float s0 = cq0 * lo[rr[0] + cc[0]] + cq1 * lo[rr[0] + cc[1]] +
             cq2 * lo[rr[0] + cc[2]] + cq3 * lo[rr[0] + cc[3]];
  float s1 = cq0 * lo[rr[1] + cc[0]] + cq1 * lo[rr[1] + cc[1]] +
             cq2 * lo[rr[1] + cc[2]] + cq3 * lo[rr[1] + cc[3]];
  float s2 = cq0 * lo[rr[2] + cc[0]] + cq1 * lo[rr[2] + cc[1]] +
             cq2 * lo[rr[2] + cc[2]] + cq3 * lo[rr[2] + cc[3]];
  float s3 = cq0 * lo[rr[3] + cc[0]] + cq1 * lo[rr[3] + cc[1]] +
             cq2 * lo[rr[3] + cc[2]] + cq3 * lo[rr[3] + cc[3]];
  float acc = cp0 * s0 + cp1 * s1 + cp2 * s2 + cp3 * s3;
  xhi[idx] = x[idx] - acc;
}

// --------------------------------------------------------------------------
// Fused quincunx FB, row flavor: z[i,j] = X[i,(j+s*i)%W];
//   lo[a,j] = (z[2a,j]+z[2a+1,j])/sqrt2 ; hi[a,j] = (z[2a+1,j]-z[(2a+2)%H,j])/sqrt2
// --------------------------------------------------------------------------
__global__ void k_qfb_r(const float* __restrict__ X, float* __restrict__ lo,
                        float* __restrict__ hi, int Hc, int Wc, int sign, int total) {
  int idx = blockIdx.x * blockDim.x + threadIdx.x;
  if (idx >= total) return;
  const int Hh = Hc >> 1;
  int j = idx % Wc;
  int t = idx / Wc;
  int a = t % Hh;
  int img = t / Hh;
  const float* xi = X + (size_t)img * Hc * Wc;
  const int i0 = 2 * a, i1 = 2 * a + 1;
  const int i2 = (i1 + 1 == Hc) ? 0 : i1 + 1;
  float z0 = xi[(size_t)i0 * Wc + imod(j + sign * i0, Wc)];
  float z1 = xi[(size_t)i1 * Wc + imod(j + sign * i1, Wc)];
  float z2 = xi[(size_t)i2 * Wc + imod(j + sign * i2, Wc)];
  const float RS2 = 0.70710678119f;
  size_t o = (size_t)img * Hh * Wc + (size_t)a * Wc + j;
  lo[o] = (z0 + z1) * RS2;
  hi[o] = (z1 - z2) * RS2;
}

// Column flavor: z[i,j] = X[(i+s*j)%H, j]
__global__ void k_qfb_c(const float* __restrict__ X, float* __restrict__ lo,
                        float* __restrict__ hi, int Hc, int Wc, int sign, int total) {
  int idx = blockIdx.x * blockDim.x + threadIdx.x;
  if (idx >= total) return;
  const int Wh = Wc >> 1;
  int b = idx % Wh;
  int t = idx / Wh;
  int i = t % Hc;
  int img = t / Hc;
  const float* xi = X + (size_t)img * Hc * Wc;
  const int j0 = 2 * b, j1 = 2 * b + 1;
  const int j2 = (j1 + 1 == Wc) ? 0 : j1 + 1;
  float z0 = xi[(size_t)imod(i + sign * j0, Hc) * Wc + j0];
  float z1 = xi[(size_t)imod(i + sign * j1, Hc) * Wc + j1];
  float z2 = xi[(size_t)imod(i + sign * j2, Hc) * Wc + j2];
  const float RS2 = 0.70710678119f;
  size_t o = (size_t)img * Hc * Wh + (size_t)i * Wh + b;
  lo[o] = (z0 + z1) * RS2;
  hi[o] = (z1 - z2) * RS2;
}

// --------------------------------------------------------------------------
// Bilinear 2x upsample (jax.image.resize: (o+0.5)/2-0.5, edge renormalize
// == clamp) + scatter into the fine output channel block.
// --------------------------------------------------------------------------
__global__ void k_resize_rows(const float* __restrict__ Y, float* __restrict__ fine,
                              int sub, int s, int C, int total) {
  int idx = blockIdx.x * blockDim.x + threadIdx.x;
  if (idx >= total) return;
  int j = idx % s;
  int t = idx / s;
  int oi = t % s;
  int img = t / s;
  int b = img / C, c = img % C;
  const int h = s >> 1;
  const float* yi = Y + (size_t)img * h * s;
  int a = oi >> 1;
  int a2 = (oi & 1) ? ((a + 1 < h) ? a + 1 : h - 1) : ((a > 0) ? a - 1 : 0);
  float val = 0.75f * yi[(size_t)a * s + j] + 0.25f * yi[(size_t)a2 * s + j];
  fine[((size_t)(b * (8 * C) + sub * C + c) * s + oi) * s + j] = val;
}

__global__ void k_resize_cols(const float* __restrict__ Y, float* __restrict__ fine,
                              int sub, int s, int C, int total) {
  int idx = blockIdx.x * blockDim.x + threadIdx.x;
  if (idx >= total) return;
  int j = idx % s;
  int t = idx / s;
  int oi = t % s;
  int img = t / s;
  int b = img / C, c = img % C;
  const int w = s >> 1;
  const float* yi = Y + (size_t)img * s * w;
  int a = j >> 1;
  int a2 = (j & 1) ? ((a + 1 < w) ? a + 1 : w - 1) : ((a > 0) ? a - 1 : 0);
  float val = 0.75f * yi[(size_t)oi * w + a] + 0.25f * yi[(size_t)oi * w + a2];
  fine[((size_t)(b * (8 * C) + sub * C + c) * s + oi) * s + j] = val;
}

// --------------------------------------------------------------------------
// 2-D DWT of coarsest xlo (64x64 -> 4 x 32x32): all four subbands share the
// same 7x7 circular window (h: taps 0..4, h1: taps 0..6, same base offset).
// Loops fully unrolled -> coefficients fold to literals.
// --------------------------------------------------------------------------
__global__ void k_wfb(const float* __restrict__ X, float* __restrict__ coarse,
                      int C, int total) {
  int idx = blockIdx.x * blockDim.x + threadIdx.x;
  if (idx >= total) return;
  int bc = idx % 32;
  int t = idx / 32;
  int a = t % 32;
  int img = t / 32;
  int b = img / C, c = img % C;
  const float* xi = X + (size_t)img * 64 * 64;
  int rows[7], cols[7];
#pragma unroll
  for (int m = 0; m < 7; ++m) {
    rows[m] = imod(2 * a - 2 + m, 64);
    cols[m] = imod(2 * bc - 2 + m, 64);
  }
  float LL = 0.f, LH = 0.f, HL = 0.f, HH = 0.f;
#pragma unroll
  for (int m = 0; m < 7; ++m) {
    const float* rp = xi + (size_t)rows[m] * 64;
#pragma unroll
    for (int k = 0; k < 7; ++k) {
      float v = rp[cols[k]];
      if (m < 5 && k < 5) LL += c_h[m] * c_h[k] * v;
      if (k < 5)          LH += c_h1[m] * c_h[k] * v;
      if (m < 5)          HL += c_h[m] * c_h1[k] * v;
      HH += c_h1[m] * c_h1[k] * v;
    }
  }
  size_t base = ((size_t)b * 4 * C + c) * 32 * 32 + (size_t)a * 32 + bc;
  const size_t blk = (size_t)C * 32 * 32;
  coarse[base]           = LL;
  coarse[base + blk]     = LH;
  coarse[base + 2 * blk] = HL;
  coarse[base + 3 * blk] = HH;
}

// --------------------------------------------------------------------------
extern "C" void kernel_launch(void* const* d_in, const int* in_sizes, int n_in,
                              void* d_out, int out_size, void* d_ws, size_t ws_size,
                              hipStream_t stream) {
  (void)in_sizes; (void)n_in; (void)out_size; (void)ws_size;
  const int B = 8, C = 16;
  const float* x = (const float*)d_in[0];
  float* out = (float*)d_out;

  const size_t N0 = (size_t)B * C * 512 * 512;   // 33,554,432
  float* P0 = (float*)d_ws;                      // xhi / a-bands
  float* P1 = P0 + N0;                           // x0x1 / y-bands
  float* LO[2] = {P1 + N0, P1 + N0 + N0 / 4};    // ping-pong xlo

  // output layout: [coarse | fine2 | fine1 | fine0]
  size_t off = (size_t)B * 4 * C * 32 * 32;
  float* fineOut[3];
  fineOut[2] = out + off; off += (size_t)B * 8 * C * 64 * 64;
  fineOut[1] = out + off; off += (size_t)B * 8 * C * 128 * 128;
  fineOut[0] = out + off;

  const float* cur = x;
  for (int l = 0; l < 3; ++l) {
    const int H = 512 >> l, W = H, H2 = H >> 1, W2 = W >> 1;
    const int NI = B * C;
    const int total = NI * H * W;
    float* xlo = LO[l & 1];

    dim3 gl(W2 / 16, H2 / 16, NI);
    k_lplo<<<gl, dim3(32), 0, stream>>>(cur, xlo, H, W);

    k_lphi<<<(total + 255) / 256, 256, 0, stream>>>(cur, xlo, P0, H, W, total);

    // DFB stage 1 (rows, sign +1): xhi -> x0, x1
    float* x0 = P1;
    float* x1 = P1 + total / 2;
    const int t2 = total / 2;
    k_qfb_r<<<(t2 + 255) / 256, 256, 0, stream>>>(P0, x0, x1, H, W, +1, t2);

    // stage 2 (cols): x0 -> a0,a1 (s=+1) ; x1 -> a2,a3 (s=-1)  (xhi dead)
    float* a0 = P0;
    float* a1 = P0 + (size_t)total / 4;
    float* a2 = P0 + (size_t)total / 2;
    float* a3 = P0 + 3 * ((size_t)total / 4);
    const int t4 = total / 4;
    k_qfb_c<<<(t4 + 255) / 256, 256, 0, stream>>>(x0, a0, a1, H2, W, +1, t4);
    k_qfb_c<<<(t4 + 255) / 256, 256, 0, stream>>>(x1, a2, a3, H2, W, -1, t4);

    // stage 3: a0/a1 row-split, a2/a3 col-split  (x0x1 dead)
    const int t8 = total / 8;
    float* y[8];
    for (int i = 0; i < 8; ++i) y[i] = P1 + (size_t)i * t8;
    k_qfb_r<<<(t8 + 255) / 256, 256, 0, stream>>>(a0, y[0], y[1], H2, W2, +1, t8);
    k_qfb_r<<<(t8 + 255) / 256, 256, 0, stream>>>(a1, y[2], y[3], H2, W2, -1, t8);
    k_qfb_c<<<(t8 + 255) / 256, 256, 0, stream>>>(a2, y[4], y[5], H2, W2, +1, t8);
    k_qfb_c<<<(t8 + 255) / 256, 256, 0, stream>>>(a3, y[6], y[7], H2, W2, -1, t8);

    // resize to square + scatter; final order [y0,y1,y2,y3,y7,y6,y5,y4]
    const int s = H2;
    const int ts = NI * s * s;
    const int nbs = (ts + 255) / 256;
    k_resize_rows<<<nbs, 256, 0, stream>>>(y[0], fineOut[l], 0, s, C, ts);
    k_resize_rows<<<nbs, 256, 0, stream>>>(y[1], fineOut[l], 1, s, C, ts);
    k_resize_rows<<<nbs, 256, 0, stream>>>(y[2], fineOut[l], 2, s, C, ts);
    k_resize_rows<<<nbs, 256, 0, stream>>>(y[3], fineOut[l], 3, s, C, ts);
    k_resize_cols<<<nbs, 256, 0, stream>>>(y[7], fineOut[l], 4, s, C, ts);
    k_resize_cols<<<nbs, 256, 0, stream>>>(y[6], fineOut[l], 5, s, C, ts);
    k_resize_cols<<<nbs, 256, 0, stream>>>(y[5], fineOut[l], 6, s, C, ts);
    k_resize_cols<<<nbs, 256, 0, stream>>>(y[4], fineOut[l], 7, s, C, ts);

    cur = xlo;
  }

  const int tw = B * C * 32 * 32;
  k_wfb<<<(tw + 255) / 256, 256, 0, stream>>>(cur, out, C, tw);
}